// DialatedRNN_90838558311003
// MI455X (gfx1250) — compile-verified
//
#include <hip/hip_runtime.h>
#include <hip/hip_bf16.h>
#include <math.h>

// Dilated RNN, D=4 independent GRU cells (H=I=O=2048) + combine + output GEMV.
// Memory-bound: ~420MB fp32 weights streamed once. GEMVs done with
// V_WMMA_F32_16X16X4_F32 (16 rows per wave, K accumulated 4 at a time,
// two independent accumulator chains to break the WMMA RAW chain).

#define Dpth 4
#define Hsz  2048
#define G3H  6144   // 3*H rows of gates per layer

typedef __attribute__((ext_vector_type(2))) float v2f;
typedef __attribute__((ext_vector_type(8))) float v8f;

// 16-row fp32 GEMV via WMMA f32 16x16x4.
// Wrow: &W[row_base][0], row stride = K floats. vlds: K-float vector in LDS.
// Returns D (16x16); column N=0 holds the 16 dot products:
//   lane 0 -> rows row_base+0..7 in acc[0..7], lane 16 -> rows +8..15.
__device__ __forceinline__ v8f gemv16_wmma(const float* __restrict__ Wrow,
                                           const float* vlds, int K) {
  const int lane = threadIdx.x & 31;
  const int half = lane >> 4;      // 0: K{0,1}, 1: K{2,3} per A-layout
  const int m    = lane & 15;      // row within the 16-row tile
  const float* wp = Wrow + (size_t)m * (size_t)K + (half << 1);
  const float* bp = vlds + (half << 1);
  v8f acc0 = {};
  v8f acc1 = {};
#pragma unroll 4
  for (int k = 0; k < K; k += 8) {
    v2f a0 = *(const v2f*)(wp + k);       // W[m][k   + 2*half ..] (global b64)
    v2f b0 = *(const v2f*)(bp + k);       // v[k   + 2*half ..]    (LDS bcast)
    v2f a1 = *(const v2f*)(wp + k + 4);   // W[m][k+4 + 2*half ..]
    v2f b1 = *(const v2f*)(bp + k + 4);   // v[k+4 + 2*half ..]
    acc0 = __builtin_amdgcn_wmma_f32_16x16x4_f32(
        false, a0, false, b0, (short)0, acc0, false, false);
    acc1 = __builtin_amdgcn_wmma_f32_16x16x4_f32(
        false, a1, false, b1, (short)0, acc1, false, false);
  }
  return acc0 + acc1;   // D tiles are additive; exact merge of the two chains
}

// ---- Kernel 1: gate GEMVs  gi = w_ih[i] @ x, gh = w_hh[i] @ memory[i] ----
// grid: D * 2 * 48 blocks of 256 threads (8 waves x 16 rows = 128 rows/block)
__global__ void __launch_bounds__(256)
gates_gemv_kernel(const float* __restrict__ x,
                  const float* __restrict__ memory,
                  const float* __restrict__ w_ih,
                  const float* __restrict__ w_hh,
                  const int*   __restrict__ step_p,
                  float* __restrict__ gi, float* __restrict__ gh) {
  __shared__ float vlds[Hsz];
  const int b     = blockIdx.x;
  const int layer = b / 96;        // 2*48 blocks per layer
  const int rem   = b % 96;
  const int which = rem / 48;      // 0 = ih (vs x), 1 = hh (vs memory)
  const int rb    = rem % 48;      // 128-row block within 6144 rows

  const int step = step_p[0];
  const int itr  = (layer + 1) * (layer + 1);
  if (step % itr != 0) return;     // uniform across block

  const float* v = which ? (memory + layer * Hsz) : x;
  const float* W = (which ? w_hh : w_ih) + (size_t)layer * G3H * Hsz;
  float* out     = (which ? gh : gi) + layer * G3H;

  // cache the K=2048 vector in LDS (coalesced float4)
  for (int t = threadIdx.x; t < Hsz / 4; t += blockDim.x)
    ((float4*)vlds)[t] = ((const float4*)v)[t];
  __syncthreads();

  const int wave     = threadIdx.x >> 5;
  const int lane     = threadIdx.x & 31;
  const int row_base = rb * 128 + wave * 16;

  v8f acc = gemv16_wmma(W + (size_t)row_base * Hsz, vlds, Hsz);

  if (lane == 0) {
#pragma unroll
    for (int j = 0; j < 8; ++j) out[row_base + j] = acc[j];
  } else if (lane == 16) {
#pragma unroll
    for (int j = 0; j < 8; ++j) out[row_base + 8 + j] = acc[j];
  }
}

// ---- Kernel 2: gate nonlinearity + residual/memory update + composite ----
// grid: H/256 blocks; thread t owns hidden index h across all 4 layers.
__global__ void __launch_bounds__(256)
combine_kernel(const float* __restrict__ residual,
               const float* __restrict__ memory,
               const float* __restrict__ b_ih,
               const float* __restrict__ b_hh,
               const float* __restrict__ gi,
               const float* __restrict__ gh,
               const int*   __restrict__ step_p,
               float* __restrict__ dout,        // [O | new_res | new_mem]
               float* __restrict__ composite) {
  const int h = blockIdx.x * blockDim.x + threadIdx.x;
  if (h >= Hsz) return;
  const int step = step_p[0];

  float comp = 0.f;
  int rank = 0;
  for (int i = 0; i < Dpth; ++i) {
    const int  itr    = (i + 1) * (i + 1);
    const bool active = (step % itr) == 0;
    const float res = residual[i * Hsz + h];
    const float mem = memory[i * Hsz + h];
    float nr, nm;
    if (active) {
      const float* gii = gi   + i * G3H;
      const float* ghh = gh   + i * G3H;
      const float* bi  = b_ih + i * G3H;
      const float* bh  = b_hh + i * G3H;
      const float i_r = gii[h]            + bi[h];
      const float i_z = gii[Hsz + h]      + bi[Hsz + h];
      const float i_n = gii[2 * Hsz + h]  + bi[2 * Hsz + h];
      const float h_r = ghh[h]            + bh[h];
      const float h_z = ghh[Hsz + h]      + bh[Hsz + h];
      const float h_n = ghh[2 * Hsz + h]  + bh[2 * Hsz + h];
      const float r = 1.f / (1.f + expf(-(i_r + h_r)));
      const float z = 1.f / (1.f + expf(-(i_z + h_z)));
      const float n = tanhf(i_n + r * h_n);
      const float h_new = (1.f - z) * n + z * mem;
      nr = 1.f / (1.f + expf(-(res + h_new)));
      nm = h_new;
      rank += 1;
      comp += h_new / (float)(rank * rank);   // positional weight 1/(j+1)^2
    } else {
      nr = res;
      nm = mem;
    }
    comp += nr / (float)itr;                  // residual weight 1/(i+1)^2
    dout[Hsz + i * Hsz + h]              = nr; // new_res
    dout[Hsz + Dpth * Hsz + i * Hsz + h] = nm; // new_mem
  }
  composite[h] = comp;
}

// ---- Kernel 3: output = out_w @ composite + out_b  (2048x2048 GEMV) ----
// grid: 2048/128 = 16 blocks of 256 threads.
__global__ void __launch_bounds__(256)
out_gemv_kernel(const float* __restrict__ out_w,
                const float* __restrict__ composite,
                const float* __restrict__ out_b,
                float* __restrict__ dout) {
  __shared__ float vlds[Hsz];
  for (int t = threadIdx.x; t < Hsz / 4; t += blockDim.x)
    ((float4*)vlds)[t] = ((const float4*)composite)[t];
  __syncthreads();

  const int wave     = threadIdx.x >> 5;
  const int lane     = threadIdx.x & 31;
  const int row_base = blockIdx.x * 128 + wave * 16;

  v8f acc = gemv16_wmma(out_w + (size_t)row_base * Hsz, vlds, Hsz);

  if (lane == 0) {
#pragma unroll
    for (int j = 0; j < 8; ++j)
      dout[row_base + j] = acc[j] + out_b[row_base + j];
  } else if (lane == 16) {
#pragma unroll
    for (int j = 0; j < 8; ++j)
      dout[row_base + 8 + j] = acc[j] + out_b[row_base + 8 + j];
  }
}

extern "C" void kernel_launch(void* const* d_in, const int* in_sizes, int n_in,
                              void* d_out, int out_size, void* d_ws, size_t ws_size,
                              hipStream_t stream) {
  const float* x        = (const float*)d_in[0];
  const float* residual = (const float*)d_in[1];
  const float* memory   = (const float*)d_in[2];
  const float* w_ih     = (const float*)d_in[3];
  const float* w_hh     = (const float*)d_in[4];
  const float* b_ih     = (const float*)d_in[5];
  const float* b_hh     = (const float*)d_in[6];
  const float* out_w    = (const float*)d_in[7];
  const float* out_b    = (const float*)d_in[8];
  const int*   step     = (const int*)d_in[9];

  float* gi        = (float*)d_ws;            // [D, 3H]
  float* gh        = gi + Dpth * G3H;         // [D, 3H]
  float* composite = gh + Dpth * G3H;         // [H]
  float* dout      = (float*)d_out;           // [O | D*H | D*H]

  // 1) gate GEMVs: D layers x {ih,hh} x 48 row-blocks of 128 rows
  gates_gemv_kernel<<<Dpth * 2 * 48, 256, 0, stream>>>(
      x, memory, w_ih, w_hh, step, gi, gh);

  // 2) gate nonlinearity + state update + composite vector
  combine_kernel<<<Hsz / 256, 256, 0, stream>>>(
      residual, memory, b_ih, b_hh, gi, gh, step, dout, composite);

  // 3) output projection GEMV
  out_gemv_kernel<<<Hsz / 128, 256, 0, stream>>>(
      out_w, composite, out_b, dout);
}